// Flux_Kernels_3195455668552
// MI455X (gfx1250) — compile-verified
//
#include <hip/hip_runtime.h>
#include <stdint.h>

// 5-point Dirichlet flux stencil on 4096x4096 fp32.
// Bandwidth-bound (128 MiB traffic @ 23.3 TB/s ~ 5.6 us). Strategy:
//  - Tensor Data Mover: one tensor_load_to_lds stages the whole 34x128 fp32
//    tile (row halos included) into LDS, tracked by TENSORcnt.
//  - Dirichlet ghost rows written straight into LDS (boundary formula ==
//    interior formula with ghost value).
//  - B128 LDS reads / global stores; each input element fetched once.

#define NXG 4096
#define NYG 4096
#define TX  128            // tile cols (contiguous dim)
#define TY  32             // tile output rows
#define SR  (TY + 2)       // staged rows incl. halo
#define NTHREADS 256

typedef uint32_t u32;
typedef __attribute__((ext_vector_type(4))) u32 u32x4;
typedef __attribute__((ext_vector_type(8))) u32 u32x8;

__global__ __launch_bounds__(NTHREADS) void flux_stencil_kernel(
    const float* __restrict__ u,
    const float* __restrict__ Deff,
    const float* __restrict__ dval,
    const float* __restrict__ sten,
    float* __restrict__ out)
{
    __shared__ float tile[SR * TX];   // 34 * 128 * 4B = 17408 B

    const int tid = threadIdx.x;
    const int j0  = blockIdx.x * TX;  // col origin
    const int i0  = blockIdx.y * TY;  // row origin

    // uniform scalars
    const float D   = Deff[0];
    const float s0  = sten[0];
    const float s1  = sten[1];
    const float dv0 = dval[0];  // x- ghost (row -1)
    const float dv1 = dval[1];  // x+ ghost (row NX)
    const float dv2 = dval[2];  // y- ghost (col -1)
    const float dv3 = dval[3];  // y+ ghost (col NY)

#if defined(__gfx1250__)
    // ---- TDM: one 2D tensor tile load stages all in-range rows ----
    if (tid < 32) {   // wave 0 only (uniform branch; TDM ignores EXEC)
        const int start = (i0 == 0) ? 1 : 0;              // first staged row TDM writes
        const int gi_lo = i0 - 1 + start;                 // first global row loaded
        const int gi_hi = (i0 + TY < NXG) ? (i0 + TY) : (NXG - 1);
        const u32 nrows = (u32)(gi_hi - gi_lo + 1);       // 34 interior, 33 at edges

        const uint64_t gaddr = (uint64_t)(uintptr_t)(u + (size_t)gi_lo * NYG + (size_t)j0);
        const u32 lds_base   = (u32)(uintptr_t)&tile[start * TX];  // WG-relative LDS byte addr

        // D# group 0: count=1 | lds_addr | global_addr[56:0] | type=2
        u32x4 g0;
        g0.x = 1u;                                  // count=1, ctrl/gather bits 0
        g0.y = lds_base;                            // bits [63:32] lds_addr
        g0.z = (u32)gaddr;                          // bits [95:64] global_addr lo
        g0.w = ((u32)(gaddr >> 32) & 0x01FFFFFFu)   // bits [120:96] global_addr hi
             | 0x80000000u;                         // bits [127:126] type=2
        // D# group 1: data_size=4B, tensor 4096x4096, tile 128 x nrows, row stride 4096
        u32x8 g1;
        g1.s0 = 0x00020000u;                        // wg_mask=0, data_size=2 (4B), flags=0
        g1.s1 = ((u32)NYG & 0xFFFFu) << 16;         // tensor_dim0 lo16 (atomic_bar_addr=0)
        g1.s2 = (((u32)NYG >> 16) & 0xFFFFu)        // tensor_dim0 hi16
              | (((u32)NXG & 0xFFFFu) << 16);       // tensor_dim1 lo16
        g1.s3 = (((u32)NXG >> 16) & 0xFFFFu)        // tensor_dim1 hi16
              | (((u32)TX) << 16);                  // tile_dim0 = 128 elements
        g1.s4 = nrows;                              // tile_dim1 = rows; tile_dim2 = 0
        g1.s5 = (u32)NYG;                           // tensor_dim0_stride lo32 (elements)
        g1.s6 = 0u;                                 // stride hi16; dim1_stride lo16
        g1.s7 = 0u;

        asm volatile("tensor_load_to_lds %0, %1" :: "s"(g0), "s"(g1) : "memory");
#if __has_builtin(__builtin_amdgcn_s_wait_tensorcnt)
        __builtin_amdgcn_s_wait_tensorcnt(0);
#else
        asm volatile("s_wait_tensorcnt 0x0" ::: "memory");
#endif
    }
    // Dirichlet ghost rows (disjoint from TDM-written rows)
    if (i0 == 0 && tid < TX / 4) {
        float* lp = &tile[tid * 4];
        lp[0] = dv0; lp[1] = dv0; lp[2] = dv0; lp[3] = dv0;
    }
    if (i0 + TY == NXG && tid < TX / 4) {
        float* lp = &tile[(SR - 1) * TX + tid * 4];
        lp[0] = dv1; lp[1] = dv1; lp[2] = dv1; lp[3] = dv1;
    }
#else
    // generic fallback staging (non-gfx1250 passes)
    for (int idx = tid; idx < SR * (TX / 4); idx += NTHREADS) {
        const int r  = idx >> 5;
        const int c4 = idx & 31;
        const int gi = i0 - 1 + r;
        float* lp = &tile[r * TX + c4 * 4];
        if (gi >= 0 && gi < NXG) {
            *(float4*)lp = *(const float4*)&u[(size_t)gi * NYG + (size_t)(j0 + c4 * 4)];
        } else {
            const float fv = (gi < 0) ? dv0 : dv1;
            lp[0] = fv; lp[1] = fv; lp[2] = fv; lp[3] = fv;
        }
    }
#endif
    __syncthreads();

    // ---- compute: each thread does a float4 of output for 4 rows ----
    const int c4 = tid & 31;     // float4 column within tile
    const int rg = tid >> 5;     // row group 0..7

#pragma unroll
    for (int m = 0; m < 4; ++m) {
        const int orow = rg + m * 8;        // 0..31 output row in tile
        const int sr   = orow + 1;          // staged row index
        const int gi   = i0 + orow;         // global row (always in range)
        const float* rowc = &tile[sr * TX];

        const float4 uc = *(const float4*)&rowc[c4 * 4];
        const float4 un = *(const float4*)&tile[(sr - 1) * TX + c4 * 4];
        const float4 us = *(const float4*)&tile[(sr + 1) * TX + c4 * 4];

        // west neighbor of element 0 / east neighbor of element 3
        float w, e;
        if (c4 == 0) {
            w = (j0 == 0) ? dv2 : u[(size_t)gi * NYG + (size_t)(j0 - 1)];
        } else {
            w = rowc[c4 * 4 - 1];
        }
        if (c4 == 31) {
            e = (j0 + TX >= NYG) ? dv3 : u[(size_t)gi * NYG + (size_t)(j0 + TX)];
        } else {
            e = rowc[c4 * 4 + 4];
        }

        // mirror reference: (s0*nb + s1*c)*D per direction, then sum l+r+t+b
        float4 o;
        {
            const float c0 = uc.x;
            o.x = ((s0 * un.x + s1 * c0) * D) + ((s0 * us.x + s1 * c0) * D)
                + ((s0 * w    + s1 * c0) * D) + ((s0 * uc.y + s1 * c0) * D);
        }
        {
            const float c1 = uc.y;
            o.y = ((s0 * un.y + s1 * c1) * D) + ((s0 * us.y + s1 * c1) * D)
                + ((s0 * uc.x + s1 * c1) * D) + ((s0 * uc.z + s1 * c1) * D);
        }
        {
            const float c2 = uc.z;
            o.z = ((s0 * un.z + s1 * c2) * D) + ((s0 * us.z + s1 * c2) * D)
                + ((s0 * uc.y + s1 * c2) * D) + ((s0 * uc.w + s1 * c2) * D);
        }
        {
            const float c3 = uc.w;
            o.w = ((s0 * un.w + s1 * c3) * D) + ((s0 * us.w + s1 * c3) * D)
                + ((s0 * uc.z + s1 * c3) * D) + ((s0 * e    + s1 * c3) * D);
        }

        *(float4*)&out[(size_t)gi * NYG + (size_t)(j0 + c4 * 4)] = o;
    }
}

extern "C" void kernel_launch(void* const* d_in, const int* in_sizes, int n_in,
                              void* d_out, int out_size, void* d_ws, size_t ws_size,
                              hipStream_t stream) {
    // inputs: 0=u_main (16M f32), 1=u_coupled (unused), 2=D_eff (1),
    //         3=dirichlet_val (4), 4=stencil (2), 5=t (unused)
    const float* u    = (const float*)d_in[0];
    const float* Deff = (const float*)d_in[2];
    const float* dval = (const float*)d_in[3];
    const float* sten = (const float*)d_in[4];
    float* out = (float*)d_out;

    dim3 grid(NYG / TX, NXG / TY);   // (32, 128)
    dim3 block(NTHREADS);
    flux_stencil_kernel<<<grid, block, 0, stream>>>(u, Deff, dval, sten, out);
}